// PV_DV_2D_ME_78125455114592
// MI455X (gfx1250) — compile-verified
//
#include <hip/hip_runtime.h>
#include <hip/hip_bf16.h>
#include <math.h>

typedef float v2f __attribute__((ext_vector_type(2)));
typedef float v8f __attribute__((ext_vector_type(8)));

#define EPSF 1e-9f

// --- workspace header (first 1024 bytes of d_ws), arrays follow ---
struct Scal {
  unsigned long long pvkey;   // argmax(lex): (f32bits<<32)|~idx
  unsigned long long topkey;  // per-iteration top-k argmax key
  unsigned long long sy, sx, sy2, sx2;   // exact integer coordinate sums
  unsigned cmaxbits;          // max(C) as monotone f32 bits (C>=0)
  int pvidx;
  float pvy, pvx, R, qden;
  int chosen[10];
};

// Branchless masked gathers: clamp index to 0 (always valid), load
// unconditionally, select.  Lets the compiler issue whole load clauses with a
// single s_wait instead of per-element EXEC branches + waits.
__device__ inline float r8val(int j, const float* __restrict__ feats, bool pv) {
  float f = feats[j < 0 ? 0 : j];
  return (pv && j >= 0 && f > 0.f) ? 1.f : 0.f;
}
__device__ inline float gath(const float* __restrict__ v, int j) {
  float f = v[j < 0 ? 0 : j];
  return j >= 0 ? f : 0.f;
}
// B matrix entry B[k][n], k=0..7 rows, columns: 0:ux 1:uy 2:ux^2 3:uy^2 4:ux*uy
// 5:(proj>0) 6:(proj<0), rest zero.  Gives m2 / M3 / Rplus / Rminus in one GEMM.
__device__ inline float bval(int k, int n, const float* __restrict__ unit8,
                             float b0, float b1) {
  float ux = unit8[2 * k], uy = unit8[2 * k + 1];
  float proj = ux * b0 + uy * b1;
  switch (n) {
    case 0: return ux;
    case 1: return uy;
    case 2: return ux * ux;
    case 3: return uy * uy;
    case 4: return ux * uy;
    case 5: return proj > 0.f ? 1.f : 0.f;
    case 6: return proj < 0.f ? 1.f : 0.f;
    default: return 0.f;
  }
}
__device__ inline unsigned long long shflxor_u64(unsigned long long v, int m) {
  unsigned lo = (unsigned)v, hi = (unsigned)(v >> 32);
  lo = __shfl_xor(lo, m, 32);
  hi = __shfl_xor(hi, m, 32);
  return ((unsigned long long)hi << 32) | lo;
}
__device__ inline unsigned long long wave_max_u64(unsigned long long v) {
  #pragma unroll
  for (int m = 16; m; m >>= 1) {
    unsigned long long o = shflxor_u64(v, m);
    v = o > v ? o : v;
  }
  return v;
}
__device__ inline unsigned wave_add_u32(unsigned v) {
  #pragma unroll
  for (int m = 16; m; m >>= 1) v += __shfl_xor(v, m, 32);
  return v;
}
__device__ inline unsigned wave_max_u32(unsigned v) {
  #pragma unroll
  for (int m = 16; m; m >>= 1) {
    unsigned o = __shfl_xor(v, m, 32);
    v = o > v ? o : v;
  }
  return v;
}

__global__ void k0_init(Scal* sc) {
  if (threadIdx.x == 0) {
    sc->pvkey = 0ull; sc->topkey = 0ull;
    sc->sy = 0ull; sc->sx = 0ull; sc->sy2 = 0ull; sc->sx2 = 0ull;
    sc->cmaxbits = 0u; sc->pvidx = 0;
    for (int j = 0; j < 10; ++j) sc->chosen[j] = -2;
  }
}

// Pass 1: batched [N x 8] @ [8 x 16] via V_WMMA_F32_16X16X4_F32 (K=8 = 2 chained
// K=4 WMMAs).  Each wave handles 2 tiles of 16 points = 32 points; A fragments
// are the gathered neighbor occupancies loaded straight into the documented
// 16x4 f32 A layout (lane -> M row, VGPR slot -> K).  D (16x16 f32, 8 VGPRs)
// is unpacked through LDS so the elementwise tail runs one-point-per-lane.
__global__ __launch_bounds__(256) void k1_moments(
    const float* __restrict__ feats, const int* __restrict__ coords,
    const int* __restrict__ nbr8, const float* __restrict__ unit8,
    const float* __restrict__ beam,
    float* __restrict__ ws_m2x, float* __restrict__ ws_m2y,
    float* __restrict__ ws_T, float* __restrict__ ws_C,
    float* __restrict__ ws_lp, Scal* sc, int N) {
  __shared__ float lds[8][7][32];
  const int tid  = threadIdx.x;
  const int w    = tid >> 5;
  const int lane = tid & 31;
  const int half = lane >> 4;
  const int mrow = lane & 15;
  const int base = blockIdx.x * 256 + w * 32;

  float be0 = beam[0], be1 = beam[1];
  float bn  = sqrtf(be0 * be0 + be1 * be1) + EPSF;
  float b0  = be0 / bn, b1 = be1 / bn;

  // B fragments (4x16 KxN layout mirrors A: lane -> N col, VGPR slot -> K)
  const int kb = half * 2;
  v2f B0, B1;
  B0.x = bval(kb + 0, mrow, unit8, b0, b1);
  B0.y = bval(kb + 1, mrow, unit8, b0, b1);
  B1.x = bval(kb + 4, mrow, unit8, b0, b1);
  B1.y = bval(kb + 5, mrow, unit8, b0, b1);

  #pragma unroll
  for (int t = 0; t < 2; ++t) {
    const int p  = base + t * 16 + mrow;
    const bool pv = p < N;
    // nbr8 rows are 32B; pairs (kb,kb+1) and (kb+4,kb+5) are int2-aligned.
    const int2* nb2 = (const int2*)(nbr8 + (size_t)(pv ? p : 0) * 8);
    int2 ja = nb2[half];        // k = kb, kb+1
    int2 jb = nb2[half + 2];    // k = kb+4, kb+5
    v2f A0, A1;
    A0.x = r8val(ja.x, feats, pv);
    A0.y = r8val(ja.y, feats, pv);
    A1.x = r8val(jb.x, feats, pv);
    A1.y = r8val(jb.y, feats, pv);
    v8f acc = {0.f, 0.f, 0.f, 0.f, 0.f, 0.f, 0.f, 0.f};
    acc = __builtin_amdgcn_wmma_f32_16x16x4_f32(false, A0, false, B0,
                                                (short)0, acc, false, false);
    acc = __builtin_amdgcn_wmma_f32_16x16x4_f32(false, A1, false, B1,
                                                (short)0, acc, false, false);
    if (mrow < 7) {                      // columns 0..6 carry results
      int mbase = t * 16 + half * 8;     // D rows M = half*8 + r
      #pragma unroll
      for (int r = 0; r < 8; ++r) lds[w][mrow][mbase + r] = acc[r];
    }
  }
  __syncthreads();

  const int p2 = base + lane;
  float mx  = lds[w][0][lane], my  = lds[w][1][lane];
  float Mxx = lds[w][2][lane], Myy = lds[w][3][lane], Mxy = lds[w][4][lane];
  float Rp  = lds[w][5][lane], Rm  = lds[w][6][lane];

  unsigned yy = 0, xx = 0, cbits = 0;
  if (p2 < N) {
    float T    = Mxx + Myy;
    float dM   = Mxx - Myy;
    float rad  = sqrtf(dM * dM + 4.f * Mxy * Mxy + 1e-20f);
    float lam2 = 0.5f * (T - rad);
    float C    = fmaxf(lam2 / (T + EPSF), 0.f);
    float mn   = sqrtf(mx * mx + my * my) + EPSF;
    float Fwd  = fmaxf(mx * b0 + my * b1, 0.f) / mn;
    float Bpv  = Rp / (Rp + Rm + EPSF);
    ws_m2x[p2] = mx;
    ws_m2y[p2] = my;
    ws_T[p2]   = T;
    ws_C[p2]   = C;
    ws_lp[p2]  = 0.001f * Bpv + 1e-6f * C + 1e-9f * Fwd;
    cbits = __float_as_uint(C);
    yy = (unsigned)coords[3 * (size_t)p2 + 1];
    xx = (unsigned)coords[3 * (size_t)p2 + 2];
  }
  unsigned sy  = wave_add_u32(yy);
  unsigned sx  = wave_add_u32(xx);
  unsigned sy2 = wave_add_u32(yy * yy);
  unsigned sx2 = wave_add_u32(xx * xx);
  unsigned cm  = wave_max_u32(cbits);
  if (lane == 0) {
    atomicAdd((unsigned long long*)&sc->sy,  (unsigned long long)sy);
    atomicAdd((unsigned long long*)&sc->sx,  (unsigned long long)sx);
    atomicAdd((unsigned long long*)&sc->sy2, (unsigned long long)sy2);
    atomicAdd((unsigned long long*)&sc->sx2, (unsigned long long)sx2);
    atomicMax(&sc->cmaxbits, cm);
  }
}

// Pass 2: divergence stencil on m2, lex score, global argmax via packed-u64 max.
__global__ __launch_bounds__(256) void k2_lex(
    const int* __restrict__ nbr8, const float* __restrict__ m2x,
    const float* __restrict__ m2y, const float* __restrict__ wsT,
    const float* __restrict__ wslp, float* __restrict__ out_lex,
    Scal* sc, int N) {
  int i = blockIdx.x * blockDim.x + threadIdx.x;
  unsigned long long key = 0ull;
  if (i < N) {
    int4 nb = *(const int4*)(nbr8 + (size_t)i * 8);   // rows 16B-aligned
    float a = gath(m2x, nb.x), b = gath(m2x, nb.y);
    float c = gath(m2y, nb.z), d = gath(m2y, nb.w);
    float div = 0.5f * (a - b) + 0.5f * (c - d);
    float Tr  = fmaxf(wsT[i], EPSF);
    float S   = fmaxf(div, 0.f) / (Tr + EPSF);
    float lex = S + wslp[i];
    out_lex[i] = lex;
    key = ((unsigned long long)__float_as_uint(lex) << 32) |
          (unsigned)(~(unsigned)i);
  }
  key = wave_max_u64(key);
  if ((threadIdx.x & 31) == 0) atomicMax(&sc->pvkey, key);
}

// Pass 3: decode pv, closed-form R_pix from exact integer sums, qden.
__global__ void k3_scalars(const int* __restrict__ coords, float* d_out,
                           Scal* sc, int N) {
  if (blockIdx.x == 0 && threadIdx.x == 0) {
    int idx = (int)(~(unsigned)sc->pvkey);
    sc->pvidx = idx;
    long long iy = coords[3 * (size_t)idx + 1];
    long long ix = coords[3 * (size_t)idx + 2];
    sc->pvy = (float)iy;
    sc->pvx = (float)ix;
    long long Sy  = (long long)sc->sy,  Sx  = (long long)sc->sx;
    long long Sy2 = (long long)sc->sy2, Sx2 = (long long)sc->sx2;
    long long S2 = (Sy2 + Sx2) - 2 * (iy * Sy + ix * Sx) +
                   (long long)N * (iy * iy + ix * ix);
    float R = (float)sqrt((double)S2 / (double)N) + EPSF;
    sc->R = R;
    sc->qden = fmaxf(__uint_as_float(sc->cmaxbits), EPSF);
    size_t o = 2 * (size_t)N;
    d_out[o]     = sc->pvy;   // pv_yx
    d_out[o + 1] = sc->pvx;
    d_out[o + 2] = R;         // R_pix
  }
}

// Pass 4: DV per point.
__global__ __launch_bounds__(256) void k4_dv(
    const int* __restrict__ coords, const float* __restrict__ m2x,
    const float* __restrict__ m2y, const float* __restrict__ wsC,
    const float* __restrict__ beam, float* __restrict__ out_dv,
    const Scal* sc, int N) {
  int i = blockIdx.x * blockDim.x + threadIdx.x;
  if (i >= N) return;
  float be0 = beam[0], be1 = beam[1];
  float bn  = sqrtf(be0 * be0 + be1 * be1) + EPSF;
  float b0  = be0 / bn, b1 = be1 / bn;
  float dy = (float)coords[3 * (size_t)i + 1] - sc->pvy;
  float dx = (float)coords[3 * (size_t)i + 2] - sc->pvx;
  float disp = fmaxf(dy * b0 + dx * b1, 0.f) / sc->R;
  float rn = sqrtf(dy * dy + dx * dx) + EPSF;
  float mx = m2x[i], my = m2y[i];
  float mn = sqrtf(mx * mx + my * my) + EPSF;
  float Hf  = fmaxf((mx * dy + my * dx) / rn, 0.f) / mn;
  float Fwd = fmaxf(mx * b0 + my * b1, 0.f) / mn;
  float C = wsC[i];
  float Q = 1.f - C / sc->qden;
  out_dv[i] = C * disp * Fwd * Hf * Q;
}

// Pass 5: 9-neighbor max pool + peak mask (written as 0.0/1.0 floats).
// Branchless: all 9 gathers issue as one clause, masked by select afterwards.
__global__ __launch_bounds__(256) void k5_peak(
    const int* __restrict__ nbr9, const float* __restrict__ dv,
    float* __restrict__ out_peak, int N) {
  int i = blockIdx.x * blockDim.x + threadIdx.x;
  if (i >= N) return;
  const int* nb = nbr9 + (size_t)i * 9;
  int   idxs[9];
  float vals[9];
  #pragma unroll
  for (int j = 0; j < 9; ++j) idxs[j] = nb[j];
  #pragma unroll
  for (int j = 0; j < 9; ++j) vals[j] = dv[idxs[j] < 0 ? 0 : idxs[j]];
  float pooled = -__builtin_inff();
  #pragma unroll
  for (int j = 0; j < 9; ++j)
    pooled = fmaxf(pooled, idxs[j] >= 0 ? vals[j] : -__builtin_inff());
  float v = dv[i];
  out_peak[i] = (v >= pooled && v > 0.f) ? 1.f : 0.f;
}

// Top-k: 10 iterations of masked global argmax (exact, deterministic).
__global__ __launch_bounds__(256) void k6_topk_pass(
    const float* __restrict__ dv, const float* __restrict__ peak,
    Scal* sc, int k, int N) {
  int i = blockIdx.x * blockDim.x + threadIdx.x;
  unsigned long long key = 0ull;
  if (i < N && peak[i] != 0.f) {
    bool taken = false;
    for (int j = 0; j < k; ++j) taken |= (sc->chosen[j] == i);
    if (!taken)
      key = ((unsigned long long)__float_as_uint(dv[i]) << 32) |
            (unsigned)(~(unsigned)i);
  }
  key = wave_max_u64(key);
  if ((threadIdx.x & 31) == 0 && key) atomicMax(&sc->topkey, key);
}

__global__ void k7_topk_collect(float* d_out, Scal* sc, int k, int N) {
  if (blockIdx.x == 0 && threadIdx.x == 0) {
    unsigned long long key = sc->topkey;
    float v; int idx;
    if (key) {
      v = __uint_as_float((unsigned)(key >> 32));
      idx = (int)(~(unsigned)key);
    } else {
      v = -__builtin_inff();
      idx = -1;
    }
    sc->chosen[k] = idx;
    sc->topkey = 0ull;
    size_t b = 3 * (size_t)N + 3;
    d_out[b + k]      = v;            // top_vals
    d_out[b + 10 + k] = (float)idx;   // top_idx (as float, single-dtype d_out)
  }
}

extern "C" void kernel_launch(void* const* d_in, const int* in_sizes, int n_in,
                              void* d_out, int out_size, void* d_ws,
                              size_t ws_size, hipStream_t stream) {
  (void)n_in; (void)out_size; (void)ws_size;
  const float* feats  = (const float*)d_in[0];
  const int*   coords = (const int*)d_in[1];
  const int*   nbr8   = (const int*)d_in[2];
  const int*   nbr9   = (const int*)d_in[3];
  const float* unit8  = (const float*)d_in[4];
  const float* beam   = (const float*)d_in[5];
  const int N = in_sizes[0];

  float* out = (float*)d_out;
  char*  ws  = (char*)d_ws;
  Scal*  sc  = (Scal*)ws;
  float* wsf = (float*)(ws + 1024);
  float* m2x  = wsf;
  float* m2y  = wsf + 1 * (size_t)N;
  float* wsT  = wsf + 2 * (size_t)N;
  float* wsC  = wsf + 3 * (size_t)N;
  float* wslp = wsf + 4 * (size_t)N;

  float* out_dv   = out;                       // DV      [0, N)
  float* out_lex  = out + (size_t)N;           // lex     [N, 2N)
  float* out_peak = out + 2 * (size_t)N + 3;   // is_peak [2N+3, 3N+3)

  const int nb = (N + 255) / 256;
  hipLaunchKernelGGL(k0_init, dim3(1), dim3(64), 0, stream, sc);
  hipLaunchKernelGGL(k1_moments, dim3(nb), dim3(256), 0, stream,
                     feats, coords, nbr8, unit8, beam,
                     m2x, m2y, wsT, wsC, wslp, sc, N);
  hipLaunchKernelGGL(k2_lex, dim3(nb), dim3(256), 0, stream,
                     nbr8, m2x, m2y, wsT, wslp, out_lex, sc, N);
  hipLaunchKernelGGL(k3_scalars, dim3(1), dim3(64), 0, stream, coords, out, sc, N);
  hipLaunchKernelGGL(k4_dv, dim3(nb), dim3(256), 0, stream,
                     coords, m2x, m2y, wsC, beam, out_dv, sc, N);
  hipLaunchKernelGGL(k5_peak, dim3(nb), dim3(256), 0, stream,
                     nbr9, out_dv, out_peak, N);
  for (int k = 0; k < 10; ++k) {
    hipLaunchKernelGGL(k6_topk_pass, dim3(nb), dim3(256), 0, stream,
                       out_dv, out_peak, sc, k, N);
    hipLaunchKernelGGL(k7_topk_collect, dim3(1), dim3(32), 0, stream,
                       out, sc, k, N);
  }
}